// OnlineLayer_70557722739217
// MI455X (gfx1250) — compile-verified
//
#include <hip/hip_runtime.h>
#include <hip/hip_bf16.h>

// OnlineLayer (Realtime-ST-GCN) fused inference step for N=1, C=64, Gamma=9, V=25.
// out[e] = relu( relu( (x1+x2+x3)[e] + sum_{s=0..7} fifo[s][e] ) + x_res[e] )
// over e in [0,1600) (the flattened 64x25 (C,V) plane).
//
// The 9-slot FIFO sum is the reference's all-ones 1x1 conv (9->1). We map the
// 12-term reduction (x1, x2, x3, fifo[0..7], zero pad) onto chained
// V_WMMA_F32_16X16X4_F32 with B = all-ones:
//   D[m,n] = sum_k A[m,k] * 1.0   (exact f32: every product is x*1.0)
// A (16x4 f32) lane layout per CDNA5 ISA 7.12.2:
//   lanes 0-15 : VGPR0 = A[M=lane, K=0], VGPR1 = A[M=lane, K=1]
//   lanes 16-31: VGPR0 = A[M=lane-16, K=2], VGPR1 = A[M=lane-16, K=3]
// D (16x16 f32) layout: VGPR r, lanes 0-15 -> D[M=r, N=lane];
//                       lanes 16-31 -> D[M=8+r, N=lane-16].
// => lane 0's c[0..7] = row sums for elements base+0..7,
//    lane 16's c[0..7] = row sums for elements base+8..15 (columns replicated).

typedef __attribute__((ext_vector_type(2))) float v2f;
typedef __attribute__((ext_vector_type(8))) float v8f;

#define NELEM 1600   // 64 * 25
#define NGROUP 100   // NELEM / 16

__device__ __forceinline__ float load_term(int k, int e,
                                           const float* __restrict__ x1,
                                           const float* __restrict__ x2,
                                           const float* __restrict__ x3,
                                           const float* __restrict__ fifo) {
  // terms: 0->x1, 1->x2, 2->x3, 3..10->fifo slots 0..7, 11 -> zero pad
  if (k >= 11) return 0.0f;
  const float* p;
  if (k == 0)      p = x1;
  else if (k == 1) p = x2;
  else if (k == 2) p = x3;
  else             p = fifo + (k - 3) * NELEM;
  return p[e];
}

__global__ __launch_bounds__(128) void onlinelayer_wmma_kernel(
    const float* __restrict__ x_res,
    const float* __restrict__ x1,
    const float* __restrict__ x2,
    const float* __restrict__ x3,
    const float* __restrict__ fifo,
    float* __restrict__ out) {
  const int lane   = threadIdx.x & 31;          // wave32
  const int wave   = blockIdx.x * (blockDim.x >> 5) + (threadIdx.x >> 5);
  const int nwaves = gridDim.x * (blockDim.x >> 5);

  const int  m    = lane & 15;                  // A-matrix row this lane feeds
  const bool hi   = lane >= 16;
  const int  kofs = hi ? 2 : 0;                 // hi half-wave supplies K=2,3

  const v2f bones = {1.0f, 1.0f};               // B = all ones (layout-agnostic)

  for (int g = wave; g < NGROUP; g += nwaves) {
    const int base = g * 16;
    const int e    = base + m;

    v8f c = {0.f, 0.f, 0.f, 0.f, 0.f, 0.f, 0.f, 0.f};

    // 3 chained K=4 WMMAs accumulate the 12 reduction terms.
#pragma unroll
    for (int stage = 0; stage < 3; ++stage) {
      const int k0 = stage * 4 + kofs;
      v2f a;
      a.x = load_term(k0,     e, x1, x2, x3, fifo);
      a.y = load_term(k0 + 1, e, x1, x2, x3, fifo);
      // 8 args: (neg_a, A, neg_b, B, c_mod, C, reuse_a, reuse_b)
      c = __builtin_amdgcn_wmma_f32_16x16x4_f32(
          false, a, false, bones, (short)0, c, false, false);
    }

    // Epilogue: lane 0 owns elements base+0..7 (c[0..7] = D[M=0..7, N=0]),
    // lane 16 owns base+8..15 (c[0..7] = D[M=8..15, N=0]).
    const int ebase = (lane == 0) ? base : ((lane == 16) ? base + 8 : -1);
    if (ebase >= 0) {
      const float4 r0 = *(const float4*)(x_res + ebase);
      const float4 r1 = *(const float4*)(x_res + ebase + 4);
      float4 o0, o1;
      o0.x = fmaxf(fmaxf(c[0], 0.f) + r0.x, 0.f);
      o0.y = fmaxf(fmaxf(c[1], 0.f) + r0.y, 0.f);
      o0.z = fmaxf(fmaxf(c[2], 0.f) + r0.z, 0.f);
      o0.w = fmaxf(fmaxf(c[3], 0.f) + r0.w, 0.f);
      o1.x = fmaxf(fmaxf(c[4], 0.f) + r1.x, 0.f);
      o1.y = fmaxf(fmaxf(c[5], 0.f) + r1.y, 0.f);
      o1.z = fmaxf(fmaxf(c[6], 0.f) + r1.z, 0.f);
      o1.w = fmaxf(fmaxf(c[7], 0.f) + r1.w, 0.f);
      *(float4*)(out + ebase)     = o0;
      *(float4*)(out + ebase + 4) = o1;
    }
  }
}

extern "C" void kernel_launch(void* const* d_in, const int* in_sizes, int n_in,
                              void* d_out, int out_size, void* d_ws, size_t ws_size,
                              hipStream_t stream) {
  // setup_inputs() order: x_res, x1, x2, x3, fifo, A1, A2, A3 (A* unused in fwd)
  const float* x_res = (const float*)d_in[0];
  const float* x1    = (const float*)d_in[1];
  const float* x2    = (const float*)d_in[2];
  const float* x3    = (const float*)d_in[3];
  const float* fifo  = (const float*)d_in[4];
  float* out = (float*)d_out;

  (void)in_sizes; (void)n_in; (void)out_size; (void)d_ws; (void)ws_size;

  // 4 blocks x 128 threads = 16 wave32s; 100 groups of 16 elements.
  onlinelayer_wmma_kernel<<<4, 128, 0, stream>>>(x_res, x1, x2, x3, fifo, out);
}